// FlashAttn_62809601737151
// MI455X (gfx1250) — compile-verified
//
#include <hip/hip_runtime.h>
#include <math.h>

// Problem constants (fixed by the reference)
#define QN      12240        // Q == HW
#define CC      256          // C
#define NHEADS  8
#define NLEVELS 4
#define NPOINTS 4
#define DDIM    32           // C / HEADS
#define MT      5            // M-tiles (16 rows each) per wave: 80 rows/wave

typedef __attribute__((ext_vector_type(2))) float v2f;
typedef __attribute__((ext_vector_type(8))) float v8f;

// ---------------------------------------------------------------------------
// GEMM: C[M x NC] = A[M x 256] * W[256 x NC] + bias[NC]
// One wave computes MT=5 stacked 16x16 tiles (80x16) via
// V_WMMA_F32_16X16X4_F32 (K-step 4), software-pipelined: the fragments for
// k+4 are fetched BEFORE the WMMAs for k are issued, so the loads of the
// next step overlap the matrix ops of the current step instead of the wave
// draining loadcnt to 0 in front of every WMMA.
// Block = 128 threads = 4 waves covering 64 consecutive output columns.
// Fragment layout per CDNA5 ISA 7.12.2:
//   A (16x4 f32, 2 VGPR): lanes 0-15 M=r K={0,1}; lanes 16-31 M=r K={2,3}
//   B (4x16 f32, 2 VGPR): lanes 0-15 N=r K={0,1}; lanes 16-31 N=r K={2,3}
//   C/D (8 VGPR): vgpr j -> row (j + 8*half), col r
// Bias is per-column; every C element of a lane shares column r, so the bias
// seeds the accumulators.
// ---------------------------------------------------------------------------
__global__ __launch_bounds__(128) void gemm_f32_wmma(
    const float* __restrict__ A, const float* __restrict__ W,
    const float* __restrict__ bias, float* __restrict__ C, int NC)
{
    const int lane = threadIdx.x & 31;
    const int wave = threadIdx.x >> 5;
    const int half = lane >> 4;       // K-pair select / M-half for C
    const int r    = lane & 15;
    const int n0   = blockIdx.x * 64 + wave * 16;
    const int m0   = blockIdx.y * (16 * MT);

    // Single base pointers; per-tile/per-k displacements are compile-time
    // constants so they fold into immediate offsets of global_load_b64.
    const float* Abase = A + (size_t)(m0 + r) * CC + 2 * half;
    const float* bcol  = W + (size_t)(2 * half) * NC + n0 + r;

    // --- prologue: fetch fragments for k = 0 ---
    v2f aF[MT], bF;
    bF.x = bcol[0];
    bF.y = bcol[NC];
#pragma unroll
    for (int t = 0; t < MT; ++t) {
        aF[t].x = Abase[(size_t)t * 16 * CC + 0];
        aF[t].y = Abase[(size_t)t * 16 * CC + 1];
    }

    const float bv = bias[n0 + r];
    v8f acc[MT];
#pragma unroll
    for (int t = 0; t < MT; ++t)
        acc[t] = (v8f){bv, bv, bv, bv, bv, bv, bv, bv};

    // --- pipelined main loop: prefetch k+4, then WMMA on k ---
#pragma unroll 3
    for (int k = 0; k < CC - 4; k += 4) {
        const int kn = k + 4;
        v2f aN[MT], bN;
        bN.x = bcol[(size_t)kn * NC];
        bN.y = bcol[(size_t)(kn + 1) * NC];
#pragma unroll
        for (int t = 0; t < MT; ++t) {
            aN[t].x = Abase[(size_t)t * 16 * CC + kn];
            aN[t].y = Abase[(size_t)t * 16 * CC + kn + 1];
        }

#pragma unroll
        for (int t = 0; t < MT; ++t) {
            acc[t] = __builtin_amdgcn_wmma_f32_16x16x4_f32(
                /*neg_a=*/false, aF[t], /*neg_b=*/false, bF,
                /*c_mod=*/(short)0, acc[t], /*reuse_a=*/false, /*reuse_b=*/false);
        }

        bF = bN;
#pragma unroll
        for (int t = 0; t < MT; ++t) aF[t] = aN[t];
    }

    // --- epilogue: last K-step ---
#pragma unroll
    for (int t = 0; t < MT; ++t) {
        acc[t] = __builtin_amdgcn_wmma_f32_16x16x4_f32(
            false, aF[t], false, bF, (short)0, acc[t], false, false);
    }

#pragma unroll
    for (int t = 0; t < MT; ++t) {
#pragma unroll
        for (int j = 0; j < 8; ++j) {
            C[(size_t)(m0 + t * 16 + 8 * half + j) * NC + n0 + r] = acc[t][j];
        }
    }
}

// ---------------------------------------------------------------------------
// Softmax over the 16 (level,point) logits of each (q, head).
// attn layout: [q][h*16 + l*4 + p], row length 128.
// ---------------------------------------------------------------------------
__global__ __launch_bounds__(256) void softmax16_kernel(float* __restrict__ attn,
                                                        int total)
{
    const int t = blockIdx.x * blockDim.x + threadIdx.x;
    if (t >= total) return;
    const int q = t >> 3;
    const int h = t & 7;
    float* p = attn + (size_t)q * (NHEADS * NLEVELS * NPOINTS) + h * 16;

    float v[16];
    float m = -INFINITY;
#pragma unroll
    for (int i = 0; i < 16; ++i) { v[i] = p[i]; m = fmaxf(m, v[i]); }
    float s = 0.f;
#pragma unroll
    for (int i = 0; i < 16; ++i) { v[i] = __expf(v[i] - m); s += v[i]; }
    const float inv = 1.f / s;
#pragma unroll
    for (int i = 0; i < 16; ++i) p[i] = v[i] * inv;
}

// ---------------------------------------------------------------------------
// Deformable bilinear sampling + attention-weighted accumulation.
// One wave per (q, head); each lane owns one of the 32 D-channels, so every
// corner sample is one coalesced 128-byte read of value[idx][h*32 .. +31].
// Output: sampled[q][h*32 + d]  (row-major q x 256, feeds final GEMM)
// ---------------------------------------------------------------------------
__global__ __launch_bounds__(256) void deform_sample_kernel(
    const float* __restrict__ value,   // [HW][256] head-major channels
    const float* __restrict__ off,     // [Q][256] = h*32 + l*8 + p*2 + axis
    const float* __restrict__ attn,    // [Q][128] = h*16 + l*4 + p (softmaxed)
    const float* __restrict__ ref,     // [Q][LEVELS][2]
    const int*   __restrict__ sp,      // [LEVELS][2] = {H, W}
    const int*   __restrict__ lsi,     // [LEVELS]
    float*       __restrict__ out)     // [Q][256]
{
    const int lane = threadIdx.x & 31;
    const int pair = blockIdx.x * 8 + (threadIdx.x >> 5); // q*8 + h
    const int q = pair >> 3;
    const int h = pair & 7;

    float acc = 0.f;

#pragma unroll
    for (int l = 0; l < NLEVELS; ++l) {
        const int   H  = sp[2 * l + 0];
        const int   W  = sp[2 * l + 1];
        const float Hf = (float)H;
        const float Wf = (float)W;
        const int   start = lsi[l];
        const float rx = ref[((size_t)q * NLEVELS + l) * 2 + 0];
        const float ry = ref[((size_t)q * NLEVELS + l) * 2 + 1];

#pragma unroll
        for (int p = 0; p < NPOINTS; ++p) {
            const size_t ob = (size_t)q * CC + h * 32 + l * 8 + p * 2;
            const float x = (rx + off[ob + 0] / Wf) * Wf - 0.5f;
            const float y = (ry + off[ob + 1] / Hf) * Hf - 0.5f;
            const float x0 = floorf(x);
            const float y0 = floorf(y);
            const float fx = x - x0;
            const float fy = y - y0;
            const int xi = (int)x0;
            const int yi = (int)y0;

            const float aw = attn[(size_t)q * 128 + h * 16 + l * 4 + p];
            const float w00 = (1.f - fy) * (1.f - fx) * aw;
            const float w01 = (1.f - fy) * fx * aw;
            const float w10 = fy * (1.f - fx) * aw;
            const float w11 = fy * fx * aw;

            if (yi >= 0 && yi < H) {
                const float* row =
                    value + ((size_t)(start + yi * W)) * CC + h * 32 + lane;
                if (xi >= 0 && xi < W)         acc += w00 * row[(size_t)xi * CC];
                if (xi + 1 >= 0 && xi + 1 < W) acc += w01 * row[(size_t)(xi + 1) * CC];
            }
            if (yi + 1 >= 0 && yi + 1 < H) {
                const float* row =
                    value + ((size_t)(start + (yi + 1) * W)) * CC + h * 32 + lane;
                if (xi >= 0 && xi < W)         acc += w10 * row[(size_t)xi * CC];
                if (xi + 1 >= 0 && xi + 1 < W) acc += w11 * row[(size_t)(xi + 1) * CC];
            }
        }
    }

    out[(size_t)q * CC + h * 32 + lane] = acc;
}

// ---------------------------------------------------------------------------
// Launcher
// ---------------------------------------------------------------------------
extern "C" void kernel_launch(void* const* d_in, const int* in_sizes, int n_in,
                              void* d_out, int out_size, void* d_ws, size_t ws_size,
                              hipStream_t stream)
{
    const float* query  = (const float*)d_in[0];
    const float* refpts = (const float*)d_in[1];
    const float* inflat = (const float*)d_in[2];
    const int*   sp     = (const int*)d_in[3];
    const int*   lsi    = (const int*)d_in[4];
    const float* W_off  = (const float*)d_in[5];
    const float* b_off  = (const float*)d_in[6];
    const float* W_attn = (const float*)d_in[7];
    const float* b_attn = (const float*)d_in[8];
    const float* W_val  = (const float*)d_in[9];
    const float* b_val  = (const float*)d_in[10];
    const float* W_out  = (const float*)d_in[11];
    const float* b_out  = (const float*)d_in[12];
    float* out = (float*)d_out;

    // Workspace partition (all f32): value | off | attn | sampled
    float* value   = (float*)d_ws;              // QN * 256
    float* off     = value + (size_t)QN * CC;   // QN * 256
    float* attn    = off   + (size_t)QN * CC;   // QN * 128
    float* sampled = attn  + (size_t)QN * 128;  // QN * 256

    const int mblocks = QN / (16 * MT); // 12240 / 80 = 153 (exact)

    // 1) value = input_flatten @ W_val + b_val   (NC = 256)
    gemm_f32_wmma<<<dim3(CC / 64, mblocks), 128, 0, stream>>>(
        inflat, W_val, b_val, value, CC);

    // 2) off = query @ W_off + b_off             (NC = 256)
    gemm_f32_wmma<<<dim3(CC / 64, mblocks), 128, 0, stream>>>(
        query, W_off, b_off, off, CC);

    // 3) attn_logits = query @ W_attn + b_attn   (NC = 128)
    gemm_f32_wmma<<<dim3(128 / 64, mblocks), 128, 0, stream>>>(
        query, W_attn, b_attn, attn, 128);

    // 4) softmax over 16 per (q, head)
    {
        const int total = QN * NHEADS;
        softmax16_kernel<<<(total + 255) / 256, 256, 0, stream>>>(attn, total);
    }

    // 5) deformable bilinear sampling -> sampled[Q][256]
    deform_sample_kernel<<<(QN * NHEADS) / 8, 256, 0, stream>>>(
        value, off, attn, refpts, sp, lsi, sampled);

    // 6) out = sampled @ W_out + b_out           (NC = 256)
    gemm_f32_wmma<<<dim3(CC / 64, mblocks), 128, 0, stream>>>(
        sampled, W_out, b_out, out, CC);
}